// RTFML_55284819034748
// MI455X (gfx1250) — compile-verified
//
#include <hip/hip_runtime.h>

typedef __attribute__((ext_vector_type(2))) float v2f;
typedef __attribute__((ext_vector_type(8))) float v8f;

#define NCROPS 10
#define BATCH  32
#define TLEN   200
#define FDIM   2048
#define KSEL   3
#define FT     512            // f-tile staged in LDS per iteration
#define NROWS  (NCROPS*BATCH) // 320

// ---------------------------------------------------------------------------
// Kernel 1: top-3 indices of dropout-masked magnitudes per batch row.
// One thread per (tensor,row): 64 threads total. Tie-break = lowest index,
// matching jax.lax.top_k stability.
// ---------------------------------------------------------------------------
__global__ __launch_bounds__(64) void rtfm_topk_kernel(
    const float* __restrict__ abnr_magn, const float* __restrict__ norm_magn,
    const float* __restrict__ abnr_u,    const float* __restrict__ norm_u,
    int* __restrict__ idx_abnr,          int* __restrict__ idx_norm) {
  int tid = threadIdx.x;
  if (tid >= 2 * BATCH) return;
  const float* magn = (tid < BATCH) ? abnr_magn : norm_magn;
  const float* u    = (tid < BATCH) ? abnr_u    : norm_u;
  int*         out  = (tid < BATCH) ? idx_abnr  : idx_norm;
  int b = (tid < BATCH) ? tid : tid - BATCH;

  int sel0 = -1, sel1 = -1;
  for (int k = 0; k < KSEL; ++k) {
    float bestv = -1.0f;   // all candidates are >= 0
    int   best  = 0;
    for (int t = 0; t < TLEN; ++t) {
      if (t == sel0 || t == sel1) continue;
      float uu = u[b * TLEN + t];
      float v  = (uu >= 0.7f) ? magn[b * TLEN + t] * (1.0f / 0.3f) : 0.0f;
      if (v > bestv) { bestv = v; best = t; }   // strict > keeps lowest index on ties
    }
    out[b * KSEL + k] = best;
    if (k == 0) sel0 = best; else if (k == 1) sel1 = best;
  }
}

// ---------------------------------------------------------------------------
// Kernel 2: for a 16-row group of mean_feat (rows = nc*32+b, row base r*T*F),
// stage the 3-gather mean into LDS coalesced, then compute the squared L2
// norms of the 16 rows as diag(X * X^T) via V_WMMA_F32_16X16X4_F32.
// For this op the A-matrix layout of X and the B-matrix layout of X^T are
// identical per-lane, so one register pair feeds both operands.
// One wave per block; EXEC all-1s as WMMA requires.
// ---------------------------------------------------------------------------
__global__ __launch_bounds__(32) void rtfm_norm_kernel(
    const float* __restrict__ feats,   // (NC, B, T, F)
    const int*   __restrict__ idx,     // (B, 3)
    float*       __restrict__ l2out) { // (320,)
  __shared__ float tile[16 * FT];      // 32 KB
  __shared__ float diag[8 * 32];       // accumulator spill for diagonal pick

  const int lane  = threadIdx.x;       // 0..31, wave32
  const int rbase = blockIdx.x * 16;   // row group base (0..304)

  v8f acc = {};

  const int m    = lane & 15;
  const int koff = (lane < 16) ? 0 : 2;

  for (int fb = 0; fb < FDIM; fb += FT) {
    // ---- stage: mean over the 3 gathered T-slices, coalesced in f ----
    for (int e = lane; e < 16 * FT; e += 32) {
      int row = e / FT;                 // 0..15 within group
      int f   = fb + (e & (FT - 1));
      int r   = rbase + row;            // == nc*32 + b
      int b   = r & 31;
      const float* base = feats + (size_t)r * TLEN * FDIM;
      int t0 = idx[b * KSEL + 0];
      int t1 = idx[b * KSEL + 1];
      int t2 = idx[b * KSEL + 2];
      float s = base[(size_t)t0 * FDIM + f]
              + base[(size_t)t1 * FDIM + f]
              + base[(size_t)t2 * FDIM + f];
      tile[e] = s * (1.0f / 3.0f);
    }
    __syncthreads();

    // ---- WMMA K-loop: acc += X_chunk * X_chunk^T  (16x16x4 f32) ----
    for (int k = 0; k < FT; k += 4) {
      v2f a;
      a.x = tile[m * FT + k + koff + 0];
      a.y = tile[m * FT + k + koff + 1];
      acc = __builtin_amdgcn_wmma_f32_16x16x4_f32(
          /*neg_a=*/false, a, /*neg_b=*/false, a,
          /*c_mod=*/(short)0, acc, /*reuse_a=*/false, /*reuse_b=*/false);
    }
    __syncthreads();
  }

  // ---- extract diagonal: D[m][m] ----
#pragma unroll
  for (int i = 0; i < 8; ++i) diag[i * 32 + lane] = acc[i];
  __syncthreads();
  if (lane < 16) {
    int   mm  = lane;
    float nsq = (mm < 8) ? diag[mm * 32 + mm]
                         : diag[(mm - 8) * 32 + (mm + 16)];
    l2out[rbase + mm] = sqrtf(nsq);
  }
}

// ---------------------------------------------------------------------------
// Kernel 3: final reductions.
// loss_rtfm = mean_{320}( (|100 - l2_abn| + l2_nor)^2 )
// loss_vls  = mean_b(-clamped_log(vls_abn)) + mean_b(-clamped_log(1-vls_nor))
// out[0] = 1e-4 * loss_rtfm ; out[1] = loss_vls
// ---------------------------------------------------------------------------
__global__ __launch_bounds__(512) void rtfm_final_kernel(
    const float* __restrict__ l2_abn, const float* __restrict__ l2_nor,
    const float* __restrict__ abnr_sls, const float* __restrict__ norm_sls,
    const int* __restrict__ idx_abnr, const int* __restrict__ idx_norm,
    float* __restrict__ out) {
  __shared__ float red[512];
  __shared__ float vred[32];
  int tid = threadIdx.x;

  float term = 0.0f;
  if (tid < NROWS) {
    float la = fabsf(100.0f - l2_abn[tid]);
    float s  = la + l2_nor[tid];
    term = s * s;
  }
  red[tid] = term;

  if (tid < BATCH) {
    int b = tid;
    float va = (abnr_sls[b * TLEN + idx_abnr[b * KSEL + 0]] +
                abnr_sls[b * TLEN + idx_abnr[b * KSEL + 1]] +
                abnr_sls[b * TLEN + idx_abnr[b * KSEL + 2]]) * (1.0f / 3.0f);
    float vn = (norm_sls[b * TLEN + idx_norm[b * KSEL + 0]] +
                norm_sls[b * TLEN + idx_norm[b * KSEL + 1]] +
                norm_sls[b * TLEN + idx_norm[b * KSEL + 2]]) * (1.0f / 3.0f);
    float lga = fmaxf(logf(va), -100.0f);         // BCE target 1, clamped log
    float lgn = fmaxf(logf(1.0f - vn), -100.0f);  // BCE target 0, clamped log
    vred[b] = -(lga + lgn);
  }
  __syncthreads();

  // tree-reduce the 512-entry RTFM terms
  for (int s = 256; s > 0; s >>= 1) {
    if (tid < s) red[tid] += red[tid + s];
    __syncthreads();
  }

  if (tid == 0) {
    out[0] = 1e-4f * (red[0] / (float)NROWS);
    float vs = 0.0f;
    for (int i = 0; i < BATCH; ++i) vs += vred[i];
    out[1] = vs / (float)BATCH;
  }
}

// ---------------------------------------------------------------------------
// Launch wiring. Input order (setup_inputs):
//  0 abnr_feat_magn (B,T) 1 norm_feat_magn (B,T)
//  2 abnr_feats (NC,B,T,F) 3 norm_feats (NC,B,T,F)
//  4 abnr_sls (B,T) 5 norm_sls (B,T) 6 abnr_drop_u (B,T) 7 norm_drop_u (B,T)
// Workspace: idxA[96] idxN[96] (int) | l2A[320] l2N[320] (float)
// ---------------------------------------------------------------------------
extern "C" void kernel_launch(void* const* d_in, const int* in_sizes, int n_in,
                              void* d_out, int out_size, void* d_ws, size_t ws_size,
                              hipStream_t stream) {
  const float* abnr_magn  = (const float*)d_in[0];
  const float* norm_magn  = (const float*)d_in[1];
  const float* abnr_feats = (const float*)d_in[2];
  const float* norm_feats = (const float*)d_in[3];
  const float* abnr_sls   = (const float*)d_in[4];
  const float* norm_sls   = (const float*)d_in[5];
  const float* abnr_u     = (const float*)d_in[6];
  const float* norm_u     = (const float*)d_in[7];

  int*   idxA = (int*)d_ws;
  int*   idxN = idxA + BATCH * KSEL;                // +96
  float* l2A  = (float*)d_ws + 2 * BATCH * KSEL;    // +192 floats
  float* l2N  = l2A + NROWS;

  rtfm_topk_kernel<<<1, 64, 0, stream>>>(abnr_magn, norm_magn, abnr_u, norm_u,
                                         idxA, idxN);

  rtfm_norm_kernel<<<NROWS / 16, 32, 0, stream>>>(abnr_feats, idxA, l2A);
  rtfm_norm_kernel<<<NROWS / 16, 32, 0, stream>>>(norm_feats, idxN, l2N);

  rtfm_final_kernel<<<1, 512, 0, stream>>>(l2A, l2N, abnr_sls, norm_sls,
                                           idxA, idxN, (float*)d_out);
}